// Sparsemax_67662914781375
// MI455X (gfx1250) — compile-verified
//
#include <hip/hip_runtime.h>
#include <stdint.h>

// ---------------------------------------------------------------------------
// Sparsemax rows, D = 8192, fp32.
// Michelot active-set iteration (exact simplex projection, no sort):
//   tau0 = (sum(z) - 1)/D ;  repeat: tau = (sum_{z>tau} z - 1)/|{z>tau}|
// One workgroup per row. Row streamed global->LDS via the Tensor Data Mover
// (tensor_load_to_lds + s_wait_tensorcnt), held in VGPRs for the iterations,
// result written with nontemporal b128 stores.
// ---------------------------------------------------------------------------

typedef unsigned int u32x4 __attribute__((ext_vector_type(4)));
typedef int          i32x8 __attribute__((ext_vector_type(8)));
typedef int          i32x4 __attribute__((ext_vector_type(4)));
typedef float        f32x4 __attribute__((ext_vector_type(4)));

#define DCOL   8192
#define BLOCK  256
#define NWAVES (BLOCK / 32)
#define VPT    (DCOL / BLOCK)   // 32 elements per thread
#define VEC    (VPT / 4)        // 8 x float4 per thread

__device__ __forceinline__ float wave_reduce_add(float x) {
#pragma unroll
  for (int off = 16; off > 0; off >>= 1) x += __shfl_xor(x, off, 32);
  return x;
}

// Returns (S, C) reduced over the whole block; result uniform across block.
__device__ __forceinline__ void block_reduce2(float s, float c,
                                              float* red_s, float* red_c,
                                              int lane, int wid,
                                              float* S_out, float* C_out) {
  s = wave_reduce_add(s);
  c = wave_reduce_add(c);
  __syncthreads();                       // protect scratch from prior reads
  if (lane == 0) { red_s[wid] = s; red_c[wid] = c; }
  __syncthreads();
  float S = 0.f, C = 0.f;
#pragma unroll
  for (int w = 0; w < NWAVES; ++w) { S += red_s[w]; C += red_c[w]; }
  *S_out = S; *C_out = C;
}

__global__ __launch_bounds__(BLOCK)
void sparsemax_michelot_kernel(const float* __restrict__ z,
                               float* __restrict__ out, int B) {
  __shared__ float row[DCOL];            // 32 KB row tile (first in LDS => offset 0)
  __shared__ float red_s[NWAVES];
  __shared__ float red_c[NWAVES];

  const int r    = blockIdx.x;
  const int tid  = threadIdx.x;
  const int lane = tid & 31;
  const int wid  = tid >> 5;

  // ---- TDM: DMA row r (8192 fp32 = 32 KB) from global into LDS ------------
  // Issued by wave 0 only (TENSORcnt is per-wave); EXEC is ignored by TDM.
  if (tid < 32) {
    const uint64_t gaddr  = (uint64_t)(uintptr_t)z + (uint64_t)r * (DCOL * 4ull);
    const unsigned ldsOff = (unsigned)(uintptr_t)row;  // flat addr low 32b == LDS offset

    u32x4 g0;
    g0[0] = 1u;                                            // count=1 (valid), user mode
    g0[1] = ldsOff;                                        // lds_addr
    g0[2] = (unsigned)(gaddr & 0xFFFFFFFFull);             // global_addr[31:0]
    g0[3] = ((unsigned)(gaddr >> 32) & 0x01FFFFFFu)        // global_addr[56:32]
          | 0x80000000u;                                   // type=2 ("image")

    i32x8 g1;
    g1[0] = 0x00020000;                  // workgroup_mask=0, data_size=2 (4 bytes)
    g1[1] = (int)((DCOL & 0xFFFF) << 16);// tensor_dim0[15:0] in bits[63:48]
    g1[2] = (int)(1u << 16);             // tensor_dim0 hi=0 | tensor_dim1=1
    g1[3] = (int)((DCOL & 0xFFFF) << 16);// tensor_dim1 hi=0 | tile_dim0=8192
    g1[4] = 1;                           // tile_dim1=1, tile_dim2=0
    g1[5] = DCOL;                        // tensor_dim0_stride[31:0]
    g1[6] = (int)((DCOL & 0xFFFF) << 16);// stride0 hi=0 | tensor_dim1_stride lo
    g1[7] = 0;                           // tensor_dim1_stride hi

    i32x4 gz4 = {0, 0, 0, 0};            // groups 2/3 unused (2-D tensor)
    i32x8 gz8 = {0, 0, 0, 0, 0, 0, 0, 0};
    __builtin_amdgcn_tensor_load_to_lds(g0, g1, gz4, gz4, gz8, 0);
    __builtin_amdgcn_s_wait_tensorcnt(0);
  }
  __syncthreads();                       // publish LDS row to all 8 waves

  // ---- Cache 32 elements/thread in VGPRs (b128 LDS reads) -----------------
  float v[VPT];
  const f32x4* rowv = (const f32x4*)row;
#pragma unroll
  for (int c = 0; c < VEC; ++c) {
    f32x4 t = rowv[c * BLOCK + tid];     // lane-contiguous 16B chunks
    v[4 * c + 0] = t.x;  v[4 * c + 1] = t.y;
    v[4 * c + 2] = t.z;  v[4 * c + 3] = t.w;
  }

  // ---- tau0 = (sum - 1) / D -----------------------------------------------
  float ps = 0.f;
#pragma unroll
  for (int k = 0; k < VPT; ++k) ps += v[k];
  float S, C;
  block_reduce2(ps, 0.f, red_s, red_c, lane, wid, &S, &C);
  float tau = (S - 1.0f) * (1.0f / (float)DCOL);

  // ---- Michelot fixed-point iteration (uniform break) ---------------------
  for (int it = 0; it < 64; ++it) {
    float s = 0.f, c = 0.f;
#pragma unroll
    for (int k = 0; k < VPT; ++k) {
      if (v[k] > tau) { s += v[k]; c += 1.f; }
    }
    block_reduce2(s, c, red_s, red_c, lane, wid, &S, &C);
    if (C < 1.0f) break;                 // paranoia guard (max stays active)
    float tauN = (S - 1.0f) / C;
    if (tauN == tau) break;              // active set stable -> converged
    tau = tauN;
  }

  // ---- out = max(z - tau, 0), streamed with nontemporal b128 stores -------
  float* outr = out + (size_t)r * DCOL;
#pragma unroll
  for (int c = 0; c < VEC; ++c) {
    const int base = (c * BLOCK + tid) * 4;
    f32x4 o;
    o.x = fmaxf(v[4 * c + 0] - tau, 0.f);
    o.y = fmaxf(v[4 * c + 1] - tau, 0.f);
    o.z = fmaxf(v[4 * c + 2] - tau, 0.f);
    o.w = fmaxf(v[4 * c + 3] - tau, 0.f);
    __builtin_nontemporal_store(o, (f32x4*)(outr + base));
  }
}

extern "C" void kernel_launch(void* const* d_in, const int* in_sizes, int n_in,
                              void* d_out, int out_size, void* d_ws, size_t ws_size,
                              hipStream_t stream) {
  (void)n_in; (void)d_ws; (void)ws_size; (void)out_size;
  const float* z = (const float*)d_in[0];
  float* out     = (float*)d_out;
  const int B    = in_sizes[0] / DCOL;   // 8192 rows
  sparsemax_michelot_kernel<<<B, BLOCK, 0, stream>>>(z, out, B);
}